// ROIPool_3d_26903675142311
// MI455X (gfx1250) — compile-verified
//
#include <hip/hip_runtime.h>
#include <stdint.h>

#define FDIM 48
#define NCH  64
#define NROI 16
// max ROI edge length is 24 (rng.integers(8,25) is high-exclusive) -> 24^3 = 13824
#define MAX_VOL 13824

typedef unsigned int v4u __attribute__((ext_vector_type(4)));
typedef int          v4i __attribute__((ext_vector_type(4)));
typedef int          v8i __attribute__((ext_vector_type(8)));

__global__ __launch_bounds__(64) void ROIPool_3d_kernel(
    const float* __restrict__ input,   // [1,64,48,48,48]
    const int*   __restrict__ rois,    // [1,16,7] : b, x0,y0,z0, x1,y1,z1
    float*       __restrict__ out)     // [16,64,4,4,4]
{
    __shared__ float smem[MAX_VOL];

    const int bid = blockIdx.x;        // n*64 + c
    const int n   = bid >> 6;
    const int c   = bid & 63;
    const int t   = threadIdx.x;       // 0..63

    const int* r  = rois + n * 7;
    const int sx = r[1], sy = r[2], sz = r[3];
    const int lx = r[4] + 1 - sx;      // lengths (ends inclusive), 8..24
    const int ly = r[5] + 1 - sy;
    const int lz = r[6] + 1 - sz;

    // Low 32 bits of a generic pointer into __shared__ == LDS byte offset.
    const unsigned lds_base = (unsigned)(unsigned long long)(const void*)&smem[0];

    // ---- Phase 1: TDM 3D-tile DMA, split along X between the two waves ----
    const int wv   = t >> 5;                 // wave id within workgroup (0/1)
    const int xh   = lx >> 1;
    const int x0w  = wv ? xh : 0;            // first X-plane this wave copies
    const int xcw  = wv ? (lx - xh) : xh;    // #X-planes this wave copies (>=4)

    const float* tile0 = input + ((((size_t)c * FDIM + sx) * FDIM + sy) * FDIM + sz);
    const uint64_t ga  = (uint64_t)(uintptr_t)tile0
                       + (uint64_t)x0w * (FDIM * FDIM * sizeof(float));
    const unsigned lds_addr = lds_base + (unsigned)(x0w * ly * lz) * 4u;

    // D# group 0: count=1 | lds_addr | global_addr[56:0] | type=2
    v4u g0 = { 1u,
               lds_addr,
               (unsigned)ga,
               (unsigned)((ga >> 32) & 0x1FFFFFFu) | (2u << 30) };

    // D# group 1: data_size=4B (enc 2); tensor_dim0/1 = lz/ly (tile==tensor:
    // no OOB); tile_dim0/1/2 = lz/ly/xcw; stride0=48, stride1=48*48 elements.
    v8i g1 = { (int)0x00020000,                                   // mask=0, data_size=2
               (int)(((unsigned)lz & 0xFFFFu) << 16),             // barrier_addr=0 | dim0.lo
               (int)((((unsigned)lz >> 16) & 0xFFFFu)
                     | (((unsigned)ly & 0xFFFFu) << 16)),         // dim0.hi | dim1.lo
               (int)((((unsigned)ly >> 16) & 0xFFFFu)
                     | (((unsigned)lz & 0xFFFFu) << 16)),         // dim1.hi | tile_dim0
               (int)(((unsigned)ly & 0xFFFFu)
                     | (((unsigned)xcw & 0xFFFFu) << 16)),        // tile_dim1 | tile_dim2
               (int)FDIM,                                         // stride0[31:0] = 48
               (int)(((FDIM * FDIM) & 0xFFFF) << 16),             // stride0.hi=0 | stride1.lo
               (int)((FDIM * FDIM) >> 16) };                      // stride1[47:16] = 0

    // D# group 2: tensor_dim2 = xcw; dim3/stride2/tile_dim3 unused.
    v4i g2 = { xcw, 0, 0, 0 };
    // D# group 3: zeros (must be supplied for 3D tensors; contents unused).
    v4i g3 = { 0, 0, 0, 0 };
    // Trailing group required by the 6-arg toolchain variant; unused for a
    // plain 3D non-gather load -> zero-filled.
    v8i g4 = { 0, 0, 0, 0, 0, 0, 0, 0 };

    __builtin_amdgcn_tensor_load_to_lds(g0, g1, g2, g3, g4, 0);
    __builtin_amdgcn_s_wait_tensorcnt(0);   // each wave waits for its own DMA
    __syncthreads();                        // publish LDS workgroup-wide

    // ---- Phase 2: one thread per output bin (i,j,k) ----
    const int i = t >> 4;
    const int j = (t >> 2) & 3;
    const int k = t & 3;

    // PyTorch adaptive bins relative to ROI start: [floor(i*L/4), ceil((i+1)*L/4))
    const int bx0 = (i * lx) >> 2,  bx1 = ((i + 1) * lx + 3) >> 2;
    const int by0 = (j * ly) >> 2,  by1 = ((j + 1) * ly + 3) >> 2;
    const int bz0 = (k * lz) >> 2,  bz1 = ((k + 1) * lz + 3) >> 2;

    float m = -3.4028234663852886e38f;      // bins never empty (L >= 8)
    for (int ix = bx0; ix < bx1; ++ix) {
        for (int iy = by0; iy < by1; ++iy) {
            const float* row = &smem[(ix * ly + iy) * lz];
            for (int iz = bz0; iz < bz1; ++iz) {
                m = fmaxf(m, row[iz]);
            }
        }
    }

    // out[n][c][i][j][k] flat = bid*64 + t
    out[(bid << 6) + t] = m;
}

extern "C" void kernel_launch(void* const* d_in, const int* in_sizes, int n_in,
                              void* d_out, int out_size, void* d_ws, size_t ws_size,
                              hipStream_t stream) {
    (void)in_sizes; (void)n_in; (void)out_size; (void)d_ws; (void)ws_size;
    const float* input = (const float*)d_in[0];
    const int*   rois  = (const int*)d_in[1];
    float*       out   = (float*)d_out;

    ROIPool_3d_kernel<<<dim3(NROI * NCH), dim3(64), 0, stream>>>(input, rois, out);
}